// GCN_pyg_24687472017556
// MI455X (gfx1250) — compile-verified
//
#include <hip/hip_runtime.h>

// GCN forward for MI455X (gfx1250, wave32).
// GEMMs use V_WMMA_F32_16X16X4_F32 (full f32 precision; total GEMM work ~10 GFLOP,
// negligible vs the edge-scatter phase, so no precision sacrifice is warranted).
// GEMM inner loop is branch-free: out-of-range A rows / B cols are index-CLAMPED
// (memory-safe) instead of masked — their contributions land only in output
// rows/cols that are never stored/read, so no EXEC manipulation is needed and
// the compiler can clause the loads between back-to-back WMMAs.
// Edge gather/scatter uses coalesced float4 bursts + global_atomic_add_f32;
// the 51 MB activation buffers fit in the 192 MB L2 so the RMW traffic stays on-chip.

typedef float v2f __attribute__((ext_vector_type(2)));
typedef float v8f __attribute__((ext_vector_type(8)));

#define NUM_GRAPHS 512
#define C_OUT 10

// ---------------- utility kernels ----------------

__global__ void fill_kernel(float* __restrict__ p, float v, int n) {
    int t = blockIdx.x * blockDim.x + threadIdx.x;
    if (t < n) p[t] = v;
}

__global__ void degree_kernel(float* __restrict__ deg, const int* __restrict__ col, int E) {
    int t = blockIdx.x * blockDim.x + threadIdx.x;
    if (t < E) atomicAdd(&deg[col[t]], 1.0f);
}

__global__ void rsqrt_kernel(float* __restrict__ deg, int n) {
    int t = blockIdx.x * blockDim.x + threadIdx.x;
    if (t < n) deg[t] = rsqrtf(deg[t]);   // deg >= 1 always (self loop), matches reference
}

__global__ void relu_kernel(float* __restrict__ p, long long n) {
    long long t = (long long)blockIdx.x * blockDim.x + threadIdx.x;
    if (t < n) { float v = p[t]; p[t] = v > 0.0f ? v : 0.0f; }
}

// ---------------- WMMA GEMM: D[N x DSTRIDE] = A[N x KDIM] @ B[KDIM x BCOLS] ----------------
// One wave computes one 16x16 output tile; fully-unrolled K-loop of 32x
// V_WMMA_F32_16X16X4_F32.
// A-matrix lane layout (16x4 f32): lane L -> row M=L&15, holds K = {2*(L>>4), 2*(L>>4)+1}.
// B-matrix (4x16 f32): lane L -> col N=L&15, same K pair. C/D: 8 VGPRs, M = r + 8*(L>>4).
template <int BCOLS, int DSTRIDE, int KDIM>
__global__ __launch_bounds__(256)
void gemm_wmma_f32(const float* __restrict__ A, const float* __restrict__ B,
                   float* __restrict__ D, int N, int ntilesN) {
    int wave = blockIdx.x * (blockDim.x >> 5) + (threadIdx.x >> 5);
    int lane = threadIdx.x & 31;
    int tilesM = (N + 15) >> 4;
    if (wave >= tilesM * ntilesN) return;      // wave-uniform: EXEC stays all-ones inside
    int tm = wave / ntilesN;
    int tn = wave - tm * ntilesN;
    int m0 = tm << 4, n0 = tn << 4;
    int half = lane >> 4;                      // 0 or 1
    int l15  = lane & 15;

    // Clamp (not mask): junk A rows only reach output rows >= N (not stored);
    // junk B cols only reach output cols >= BCOLS (never read downstream).
    int row  = m0 + l15;   if (row  >= N)     row  = N - 1;
    int bcol = n0 + l15;   if (bcol >= BCOLS) bcol = BCOLS - 1;

    const float* __restrict__ Arow = A + (size_t)row * KDIM + half * 2;  // 8B-aligned pairs
    const float* __restrict__ Bcol = B + (size_t)(half * 2) * BCOLS + bcol;

    v8f acc = {0.f, 0.f, 0.f, 0.f, 0.f, 0.f, 0.f, 0.f};
#pragma unroll
    for (int k0 = 0; k0 < KDIM; k0 += 4) {
        v2f a = *(const v2f*)(Arow + k0);
        v2f b;
        b[0] = Bcol[(size_t)k0 * BCOLS];
        b[1] = Bcol[(size_t)(k0 + 1) * BCOLS];
        // (neg_a, A, neg_b, B, c_mod, C, reuse_a, reuse_b)
        acc = __builtin_amdgcn_wmma_f32_16x16x4_f32(false, a, false, b,
                                                    (short)0, acc, false, false);
    }
#pragma unroll
    for (int r = 0; r < 8; ++r) {
        int m = m0 + r + half * 8;
        if (m < N) D[(size_t)m * DSTRIDE + n0 + l15] = acc[r];
    }
}

// ---------------- self-loop + bias init: out[i,c] = lin[i,c]*dis[i]^2 + bias[c] ----------------
template <int F>
__global__ void init_agg_kernel(const float* __restrict__ lin, const float* __restrict__ dis,
                                const float* __restrict__ bias, int biasLen,
                                float* __restrict__ out, int N) {
    long long t = (long long)blockIdx.x * blockDim.x + threadIdx.x;
    if (t >= (long long)N * F) return;
    int i = (int)(t / F);
    int c = (int)(t - (long long)i * F);
    float d  = dis[i];
    float bv = (c < biasLen) ? bias[c] : 0.0f;
    out[t] = lin[t] * d * d + bv;
}

// ---------------- edge scatter: out[col] += lin[row] * dis[row]*dis[col] ----------------
// F/4 lanes per edge, each moving a float4 -> 512B (F=128) coalesced burst per edge.
template <int F>
__global__ void scatter_kernel(const float* __restrict__ lin, const float* __restrict__ dis,
                               const int* __restrict__ row, const int* __restrict__ col,
                               float* __restrict__ out, int E) {
    const int LPE = F / 4;
    long long t = (long long)blockIdx.x * blockDim.x + threadIdx.x;
    long long e = t / LPE;
    int l = (int)(t - e * LPE);
    if (e >= E) return;
    int r = row[e];
    int c = col[e];
    float s = dis[r] * dis[c];
    const float4 v = *(const float4*)(lin + (size_t)r * F + l * 4);
    float* o = out + (size_t)c * F + l * 4;
    atomicAdd(o + 0, v.x * s);
    atomicAdd(o + 1, v.y * s);
    atomicAdd(o + 2, v.z * s);
    atomicAdd(o + 3, v.w * s);
}

// ---------------- global mean pool ----------------
__global__ void pool_kernel(const float* __restrict__ h /* N x 16 padded */,
                            const int* __restrict__ batch,
                            float* __restrict__ out, float* __restrict__ cnt, int N) {
    long long t = (long long)blockIdx.x * blockDim.x + threadIdx.x;
    if (t >= (long long)N * 16) return;
    int i = (int)(t >> 4);
    int c = (int)(t & 15);
    int g = batch[i];
    if (c < C_OUT) atomicAdd(&out[(size_t)g * C_OUT + c], h[t]);
    if (c == 0)    atomicAdd(&cnt[g], 1.0f);
}

__global__ void div_kernel(float* __restrict__ out, const float* __restrict__ cnt) {
    int t = blockIdx.x * blockDim.x + threadIdx.x;
    if (t < NUM_GRAPHS * C_OUT) out[t] /= fmaxf(cnt[t / C_OUT], 1.0f);
}

// ---------------- host launch ----------------

static inline int cdiv_ll(long long a, long long b) { return (int)((a + b - 1) / b); }

extern "C" void kernel_launch(void* const* d_in, const int* in_sizes, int n_in,
                              void* d_out, int out_size, void* d_ws, size_t ws_size,
                              hipStream_t stream) {
    const float* x     = (const float*)d_in[0];
    const int*   ei    = (const int*)d_in[1];
    const int*   batch = (const int*)d_in[2];
    const float* W0 = (const float*)d_in[3];  const float* b0 = (const float*)d_in[4];
    const float* W1 = (const float*)d_in[5];  const float* b1 = (const float*)d_in[6];
    const float* W2 = (const float*)d_in[7];  const float* b2 = (const float*)d_in[8];
    const float* W3 = (const float*)d_in[9];  const float* b3 = (const float*)d_in[10];

    const int N = in_sizes[2];          // batch vector length = #nodes
    const int E = in_sizes[1] / 2;      // edge_index is (2, E) flattened
    const int H = 128;
    const int* rowi = ei;               // edge_index[0] = source
    const int* coli = ei + E;           // edge_index[1] = target

    float* out = (float*)d_out;
    float* ws  = (float*)d_ws;
    // workspace layout (floats): dis[N] | bufA[N*128] | bufB[N*128] | cnt[512]
    float* dis  = ws;
    float* bufA = ws + (((size_t)N + 255) & ~(size_t)255);
    float* bufB = bufA + (size_t)N * H;
    float* cnt  = bufB + (size_t)N * H;

    const int TB = 256;

    // --- degrees & normalization: deg = 1 (self loop) + #in-edges; dis = rsqrt(deg)
    fill_kernel<<<cdiv_ll(N, TB), TB, 0, stream>>>(dis, 1.0f, N);
    degree_kernel<<<cdiv_ll(E, TB), TB, 0, stream>>>(dis, coli, E);
    rsqrt_kernel<<<cdiv_ll(N, TB), TB, 0, stream>>>(dis, N);

    const int tilesM     = (N + 15) >> 4;
    const int gemmBlkH   = cdiv_ll((long long)tilesM * 8, TB / 32);  // 8 N-tiles for H=128
    const int gemmBlkOut = cdiv_ll((long long)tilesM * 1, TB / 32);  // 1 N-tile for C=10->16
    const long long NH   = (long long)N * H;

    // --- layer 1: x -> bufA (GEMM) -> bufB (aggregate) ; ReLU
    gemm_wmma_f32<128, 128, 128><<<gemmBlkH, TB, 0, stream>>>(x, W0, bufA, N, 8);
    init_agg_kernel<128><<<cdiv_ll(NH, TB), TB, 0, stream>>>(bufA, dis, b0, 128, bufB, N);
    scatter_kernel<128><<<cdiv_ll((long long)E * 32, TB), TB, 0, stream>>>(bufA, dis, rowi, coli, bufB, E);
    relu_kernel<<<cdiv_ll(NH, TB), TB, 0, stream>>>(bufB, NH);

    // --- layer 2
    gemm_wmma_f32<128, 128, 128><<<gemmBlkH, TB, 0, stream>>>(bufB, W1, bufA, N, 8);
    init_agg_kernel<128><<<cdiv_ll(NH, TB), TB, 0, stream>>>(bufA, dis, b1, 128, bufB, N);
    scatter_kernel<128><<<cdiv_ll((long long)E * 32, TB), TB, 0, stream>>>(bufA, dis, rowi, coli, bufB, E);
    relu_kernel<<<cdiv_ll(NH, TB), TB, 0, stream>>>(bufB, NH);

    // --- layer 3
    gemm_wmma_f32<128, 128, 128><<<gemmBlkH, TB, 0, stream>>>(bufB, W2, bufA, N, 8);
    init_agg_kernel<128><<<cdiv_ll(NH, TB), TB, 0, stream>>>(bufA, dis, b2, 128, bufB, N);
    scatter_kernel<128><<<cdiv_ll((long long)E * 32, TB), TB, 0, stream>>>(bufA, dis, rowi, coli, bufB, E);
    relu_kernel<<<cdiv_ll(NH, TB), TB, 0, stream>>>(bufB, NH);

    // --- output layer: C=10 padded to one 16-wide tile (bufA/bufB reused as N x 16)
    gemm_wmma_f32<10, 16, 128><<<gemmBlkOut, TB, 0, stream>>>(bufB, W3, bufA, N, 1);
    init_agg_kernel<16><<<cdiv_ll((long long)N * 16, TB), TB, 0, stream>>>(bufA, dis, b3, C_OUT, bufB, N);
    scatter_kernel<16><<<cdiv_ll((long long)E * 4, TB), TB, 0, stream>>>(bufA, dis, rowi, coli, bufB, E);

    // --- global mean pool (zero accumulators every call: harness replays the graph)
    fill_kernel<<<cdiv_ll(NUM_GRAPHS * C_OUT, TB), TB, 0, stream>>>(out, 0.0f, NUM_GRAPHS * C_OUT);
    fill_kernel<<<cdiv_ll(NUM_GRAPHS, TB), TB, 0, stream>>>(cnt, 0.0f, NUM_GRAPHS);
    pool_kernel<<<cdiv_ll((long long)N * 16, TB), TB, 0, stream>>>(bufB, batch, out, cnt, N);
    div_kernel<<<cdiv_ll(NUM_GRAPHS * C_OUT, TB), TB, 0, stream>>>(out, cnt);
}